// GRN_GAT_Encoder_3470333575725
// MI455X (gfx1250) — compile-verified
//
#include <hip/hip_runtime.h>
#include <hip/hip_bf16.h>
#include <math.h>

#define NN 20000
#define NE 320000
#define FIN 256
#define FE 32
#define HIDC 128
#define NH 4

typedef __attribute__((ext_vector_type(2))) float v2f;
typedef __attribute__((ext_vector_type(8))) float v8f;

// ---------- order-preserving float<->int encoding for atomicMax-based segment max ----------
__device__ __forceinline__ int enc_f32(float f) {
  int i = __float_as_int(f);
  return (i >= 0) ? i : (i ^ 0x7FFFFFFF);
}
__device__ __forceinline__ float dec_f32(int i) {
  return __int_as_float((i >= 0) ? i : (i ^ 0x7FFFFFFF));
}

__global__ void fill_i32(int* __restrict__ p, int v, int n) {
  int i = blockIdx.x * blockDim.x + threadIdx.x;
  if (i < n) p[i] = v;
}

// ---------- fp32 WMMA GEMM: C[M,N] = A[M,K] @ B[K,N], row-major ---------------------------
// One wave computes a 16x64 output tile: 4 accumulators share one A fragment per K-step
// (4x less A traffic, 4 v_wmma per A load). Requires N % 64 == 0, M % 16 == 0, K % 4 == 0.
// A frag (16x4 f32): lanes 0-15 hold K={0,1}; lanes 16-31 hold K={2,3}; M = lane%16.
// C/D layout: VGPR r -> M = r + (lane<16 ? 0 : 8), N = lane%16.
__global__ void wmma_gemm_f32(const float* __restrict__ A, const float* __restrict__ B,
                              float* __restrict__ C, int M, int N, int K) {
  const int lane = threadIdx.x & 31;
  const int wave = blockIdx.x * (blockDim.x >> 5) + (threadIdx.x >> 5);
  const int tilesN = N >> 6;  // 64-wide tiles
  const int totalTiles = (M >> 4) * tilesN;
  if (wave >= totalTiles) return;  // uniform per wave -> EXEC stays all-ones for WMMA
  const int tm = wave / tilesN, tn = wave % tilesN;
  const int row0 = tm << 4, col0 = tn << 6;
  const int l15 = lane & 15;
  const int kOff = (lane < 16) ? 0 : 2;
  const float* __restrict__ aRow = A + (size_t)(row0 + l15) * K + kOff;
  const float* __restrict__ bCol = B + (size_t)kOff * N + (col0 + l15);
  v8f acc0 = {0.f, 0.f, 0.f, 0.f, 0.f, 0.f, 0.f, 0.f};
  v8f acc1 = acc0, acc2 = acc0, acc3 = acc0;
  for (int k0 = 0; k0 < K; k0 += 4) {
    v2f a;
    a.x = aRow[k0];
    a.y = aRow[k0 + 1];
    const float* b0 = bCol + (size_t)k0 * N;
    const float* b1 = bCol + (size_t)(k0 + 1) * N;
    v2f bf0, bf1, bf2, bf3;
    bf0.x = b0[0];  bf0.y = b1[0];
    bf1.x = b0[16]; bf1.y = b1[16];
    bf2.x = b0[32]; bf2.y = b1[32];
    bf3.x = b0[48]; bf3.y = b1[48];
    acc0 = __builtin_amdgcn_wmma_f32_16x16x4_f32(false, a, false, bf0, (short)0, acc0, false, false);
    acc1 = __builtin_amdgcn_wmma_f32_16x16x4_f32(false, a, false, bf1, (short)0, acc1, false, false);
    acc2 = __builtin_amdgcn_wmma_f32_16x16x4_f32(false, a, false, bf2, (short)0, acc2, false, false);
    acc3 = __builtin_amdgcn_wmma_f32_16x16x4_f32(false, a, false, bf3, (short)0, acc3, false, false);
  }
  float* __restrict__ cPtr =
      C + (size_t)(row0 + ((lane < 16) ? 0 : 8)) * N + col0 + l15;
#pragma unroll
  for (int r = 0; r < 8; ++r) {
    cPtr[(size_t)r * N + 0]  = acc0[r];
    cPtr[(size_t)r * N + 16] = acc1[r];
    cPtr[(size_t)r * N + 32] = acc2[r];
    cPtr[(size_t)r * N + 48] = acc3[r];
  }
}

// ---------- per-node attention terms: sn[n,h]=<h[n,h,:],aS[h,:]>, dn likewise -------------
__global__ void node_alpha(const float* __restrict__ h, const float* __restrict__ aS,
                           const float* __restrict__ aD, float* __restrict__ sn,
                           float* __restrict__ dn, int n, int H, int C) {
  int idx = blockIdx.x * blockDim.x + threadIdx.x;
  if (idx >= n * H) return;
  int node = idx / H, hh = idx % H;
  const float* hp = h + (size_t)node * H * C + (size_t)hh * C;
  const float* ap = aS + (size_t)hh * C;
  const float* bp = aD + (size_t)hh * C;
  float sa = 0.f, sb = 0.f;
  for (int c = 0; c < C; ++c) { float v = hp[c]; sa += v * ap[c]; sb += v * bp[c]; }
  sn[idx] = sa;
  dn[idx] = sb;
}

// ---------- fold We against ae: Wea[f,h] = sum_c We[f, h*C+c] * ae[h,c] -------------------
__global__ void fold_edge_weight(const float* __restrict__ We, const float* __restrict__ ae,
                                 float* __restrict__ Wea, int F, int H, int C) {
  int idx = blockIdx.x * blockDim.x + threadIdx.x;
  if (idx >= F * H) return;
  int f = idx / H, hh = idx % H;
  const float* wp = We + (size_t)f * H * C + (size_t)hh * C;
  const float* ap = ae + (size_t)hh * C;
  float s = 0.f;
  for (int c = 0; c < C; ++c) s += wp[c] * ap[c];
  Wea[idx] = s;
}

// ---------- per-edge attr term: hea[e,h] = sum_f ea[e,f] * Wea[f,h] -----------------------
__global__ void edge_he_alpha(const float* __restrict__ ea, const float* __restrict__ Wea,
                              float* __restrict__ hea, int E, int F, int H) {
  int idx = blockIdx.x * blockDim.x + threadIdx.x;
  if (idx >= E * H) return;
  int e = idx / H, hh = idx % H;
  const float* ep = ea + (size_t)e * F;
  float s = 0.f;
  for (int f = 0; f < F; ++f) s += ep[f] * Wea[f * H + hh];
  hea[idx] = s;
}

// ---------- pass A: leaky-relu logits + segment max (via int atomicMax) -------------------
__global__ void edge_logits(const float* __restrict__ asn, const float* __restrict__ adn,
                            const float* __restrict__ hea, const int* __restrict__ ei,
                            int E, int H, float* __restrict__ alpha,
                            int* __restrict__ segmax) {
  int idx = blockIdx.x * blockDim.x + threadIdx.x;
  if (idx >= 2 * E * H) return;
  int e2 = idx / H, hh = idx % H;
  int e = (e2 < E) ? e2 : e2 - E;
  int s = (e2 < E) ? ei[e] : ei[E + e];
  int d = (e2 < E) ? ei[E + e] : ei[e];
  float v = asn[s * H + hh] + adn[d * H + hh] + hea[e * H + hh];
  v = (v > 0.f) ? v : 0.2f * v;  // leaky_relu, slope 0.2
  alpha[idx] = v;
  atomicMax(&segmax[d * H + hh], enc_f32(v));
}

// ---------- pass B: exp(logit - max) + segment sum ----------------------------------------
__global__ void edge_exp(const int* __restrict__ ei, int E, int H,
                         float* __restrict__ alpha, const int* __restrict__ segmax,
                         float* __restrict__ segsum) {
  int idx = blockIdx.x * blockDim.x + threadIdx.x;
  if (idx >= 2 * E * H) return;
  int e2 = idx / H, hh = idx % H;
  int e = (e2 < E) ? e2 : e2 - E;
  int d = (e2 < E) ? ei[E + e] : ei[e];
  float m = dec_f32(segmax[d * H + hh]);
  if (!isfinite(m)) m = 0.f;
  float w = __expf(alpha[idx] - m);
  alpha[idx] = w;
  atomicAdd(&segsum[d * H + hh], w);
}

// ---------- pass C: out[dst] += h[src] * (w / segsum[dst]) ; one block per edge -----------
__global__ void edge_aggregate(const float* __restrict__ h, const float* __restrict__ alpha,
                               const float* __restrict__ segsum, const int* __restrict__ ei,
                               int E, int H, int C, float* __restrict__ out) {
  const int e2 = blockIdx.x;
  const int e = (e2 < E) ? e2 : e2 - E;
  const int s = (e2 < E) ? ei[e] : ei[E + e];
  const int d = (e2 < E) ? ei[E + e] : ei[e];
  const int j = threadIdx.x << 2;  // 4 floats per thread
  const int hh = j / C;
  const float w = alpha[(size_t)e2 * H + hh];
  const float ss = segsum[(size_t)d * H + hh];
  const float coeff = (ss > 0.f) ? (w / ss) : 0.f;
  const float4 hv = *(const float4*)(h + (size_t)s * (H * C) + j);
  float* o = out + (size_t)d * (H * C) + j;
  atomicAdd(o + 0, hv.x * coeff);
  atomicAdd(o + 1, hv.y * coeff);
  atomicAdd(o + 2, hv.z * coeff);
  atomicAdd(o + 3, hv.w * coeff);
}

__global__ void bias_elu(const float* __restrict__ x, const float* __restrict__ b,
                         float* __restrict__ y, int total, int ld) {
  int i = blockIdx.x * blockDim.x + threadIdx.x;
  if (i >= total) return;
  float v = x[i] + b[i % ld];
  y[i] = (v > 0.f) ? v : (__expf(v) - 1.f);
}

__global__ void bias_prelu(float* __restrict__ x, const float* __restrict__ b,
                           const float* __restrict__ a, int total, int ld) {
  int i = blockIdx.x * blockDim.x + threadIdx.x;
  if (i >= total) return;
  float v = x[i] + b[i % ld];
  x[i] = (v > 0.f) ? v : a[0] * v;
}

__global__ void bias_out(const float* __restrict__ x, const float* __restrict__ b,
                         float* __restrict__ y, int total, int ld) {
  int i = blockIdx.x * blockDim.x + threadIdx.x;
  if (i >= total) return;
  y[i] = x[i] + b[i % ld];
}

// column sums of [rows,128] with per-thread accumulation then one atomic each -------------
__global__ void colsum(const float* __restrict__ x, float* __restrict__ gsum, int rows) {
  int t = blockIdx.x * blockDim.x + threadIdx.x;
  int col = t & 127;
  int r0 = t >> 7;
  int stride = (gridDim.x * blockDim.x) >> 7;
  float acc = 0.f;
  for (int r = r0; r < rows; r += stride) acc += x[(size_t)r * 128 + col];
  atomicAdd(&gsum[col], acc);
}

__global__ void finalize_g(const float* __restrict__ gsum, float* __restrict__ gout,
                           int rows) {
  int j = threadIdx.x;
  if (j < 128) {
    float m = gsum[j] / (float)rows;
    gout[j] = 1.f / (1.f + __expf(-m));
  }
}

static inline int cdiv(long long a, long long b) { return (int)((a + b - 1) / b); }

extern "C" void kernel_launch(void* const* d_in, const int* in_sizes, int n_in,
                              void* d_out, int out_size, void* d_ws, size_t ws_size,
                              hipStream_t stream) {
  const float* x   = (const float*)d_in[0];
  const int*   ei  = (const int*)d_in[1];   // [2, E] flat
  const float* ea  = (const float*)d_in[2]; // [E, 32]
  const float* W1  = (const float*)d_in[3];
  const float* We1 = (const float*)d_in[4];
  const float* as1 = (const float*)d_in[5];
  const float* ad1 = (const float*)d_in[6];
  const float* ae1 = (const float*)d_in[7];
  const float* b1  = (const float*)d_in[8];
  const float* W2  = (const float*)d_in[9];
  const float* We2 = (const float*)d_in[10];
  const float* as2 = (const float*)d_in[11];
  const float* ad2 = (const float*)d_in[12];
  const float* ae2 = (const float*)d_in[13];
  const float* b2  = (const float*)d_in[14];
  const float* P1  = (const float*)d_in[15];
  const float* pb1 = (const float*)d_in[16];
  const float* pa  = (const float*)d_in[17];
  const float* P2  = (const float*)d_in[18];
  const float* pb2 = (const float*)d_in[19];
  float* out = (float*)d_out;

  // ---- workspace layout (floats) ----
  float* ws = (float*)d_ws;
  size_t o = 0;
  float* h1    = ws + o; o += (size_t)NN * 512;   // also holds elu(out1+b1) afterwards
  float* out1  = ws + o; o += (size_t)NN * 512;
  float* asn1  = ws + o; o += (size_t)NN * NH;
  float* adn1  = ws + o; o += (size_t)NN * NH;
  float* wea1  = ws + o; o += (size_t)FE * NH;
  float* hea1  = ws + o; o += (size_t)NE * NH;
  float* alp1  = ws + o; o += (size_t)2 * NE * NH;
  int*   smax1 = (int*)(ws + o); o += (size_t)NN * NH;
  float* ssum1 = ws + o; o += (size_t)NN * NH;
  float* h2    = ws + o; o += (size_t)NN * HIDC;  // also holds elu(out2+b2)
  float* out2  = ws + o; o += (size_t)NN * HIDC;
  float* asn2  = ws + o; o += (size_t)NN;
  float* adn2  = ws + o; o += (size_t)NN;
  float* wea2  = ws + o; o += (size_t)FE;
  float* hea2  = ws + o; o += (size_t)NE;
  float* alp2  = ws + o; o += (size_t)2 * NE;
  int*   smax2 = (int*)(ws + o); o += (size_t)NN;
  float* ssum2 = ws + o; o += (size_t)NN;
  float* zbuf  = ws + o; o += (size_t)NN * HIDC;
  float* hpbuf = ws + o; o += (size_t)NN * HIDC;
  float* gsum  = ws + o; o += 128;

  const int E2 = 2 * NE;

  // ---- init accumulators ----
  fill_i32<<<cdiv((size_t)NN * 512, 256), 256, 0, stream>>>((int*)out1, 0, NN * 512);
  fill_i32<<<cdiv(NN * NH, 256), 256, 0, stream>>>((int*)ssum1, 0, NN * NH);
  fill_i32<<<cdiv(NN * NH, 256), 256, 0, stream>>>(smax1, (int)0x80000000, NN * NH);
  fill_i32<<<cdiv(NN * HIDC, 256), 256, 0, stream>>>((int*)out2, 0, NN * HIDC);
  fill_i32<<<cdiv(NN, 256), 256, 0, stream>>>((int*)ssum2, 0, NN);
  fill_i32<<<cdiv(NN, 256), 256, 0, stream>>>(smax2, (int)0x80000000, NN);
  fill_i32<<<1, 128, 0, stream>>>((int*)gsum, 0, 128);

  // ======== Layer 1 ========
  // h1 = x @ W1  (20000x256 @ 256x512) : 10000 16x64 wave-tiles, 4 waves/block
  wmma_gemm_f32<<<cdiv((NN / 16) * (512 / 64), 4), 128, 0, stream>>>(x, W1, h1, NN, 512, FIN);
  node_alpha<<<cdiv(NN * NH, 256), 256, 0, stream>>>(h1, as1, ad1, asn1, adn1, NN, NH, HIDC);
  fold_edge_weight<<<1, FE * NH, 0, stream>>>(We1, ae1, wea1, FE, NH, HIDC);
  edge_he_alpha<<<cdiv((size_t)NE * NH, 256), 256, 0, stream>>>(ea, wea1, hea1, NE, FE, NH);
  edge_logits<<<cdiv((size_t)E2 * NH, 256), 256, 0, stream>>>(asn1, adn1, hea1, ei, NE, NH,
                                                              alp1, smax1);
  edge_exp<<<cdiv((size_t)E2 * NH, 256), 256, 0, stream>>>(ei, NE, NH, alp1, smax1, ssum1);
  edge_aggregate<<<E2, (NH * HIDC) / 4, 0, stream>>>(h1, alp1, ssum1, ei, NE, NH, HIDC, out1);
  bias_elu<<<cdiv((size_t)NN * 512, 256), 256, 0, stream>>>(out1, b1, h1, NN * 512, 512);

  // ======== Layer 2 (H=1) ========
  wmma_gemm_f32<<<cdiv((NN / 16) * (HIDC / 64), 4), 128, 0, stream>>>(h1, W2, h2, NN, HIDC, 512);
  node_alpha<<<cdiv(NN, 256), 256, 0, stream>>>(h2, as2, ad2, asn2, adn2, NN, 1, HIDC);
  fold_edge_weight<<<1, FE, 0, stream>>>(We2, ae2, wea2, FE, 1, HIDC);
  edge_he_alpha<<<cdiv(NE, 256), 256, 0, stream>>>(ea, wea2, hea2, NE, FE, 1);
  edge_logits<<<cdiv(E2, 256), 256, 0, stream>>>(asn2, adn2, hea2, ei, NE, 1, alp2, smax2);
  edge_exp<<<cdiv(E2, 256), 256, 0, stream>>>(ei, NE, 1, alp2, smax2, ssum2);
  edge_aggregate<<<E2, HIDC / 4, 0, stream>>>(h2, alp2, ssum2, ei, NE, 1, HIDC, out2);
  bias_elu<<<cdiv((size_t)NN * HIDC, 256), 256, 0, stream>>>(out2, b2, h2, NN * HIDC, HIDC);

  // ======== Projection head ========
  wmma_gemm_f32<<<cdiv((NN / 16) * (HIDC / 64), 4), 128, 0, stream>>>(h2, P1, zbuf, NN, HIDC, HIDC);
  bias_prelu<<<cdiv((size_t)NN * HIDC, 256), 256, 0, stream>>>(zbuf, pb1, pa, NN * HIDC, HIDC);
  wmma_gemm_f32<<<cdiv((NN / 16) * (HIDC / 64), 4), 128, 0, stream>>>(zbuf, P2, hpbuf, NN, HIDC, HIDC);
  bias_out<<<cdiv((size_t)NN * HIDC, 256), 256, 0, stream>>>(hpbuf, pb2, out, NN * HIDC, HIDC);

  // g = sigmoid(mean over nodes)
  colsum<<<128, 256, 0, stream>>>(out, gsum, NN);
  finalize_g<<<1, 128, 0, stream>>>(gsum, out + (size_t)NN * HIDC, NN);
}